// AttentionBlock_87479893885507
// MI455X (gfx1250) — compile-verified
//
#include <hip/hip_runtime.h>
#include <hip/hip_bf16.h>
#include <math.h>

typedef __attribute__((ext_vector_type(2))) float v2f;
typedef __attribute__((ext_vector_type(8))) float v8f;

#define CCH   128          // channels
#define DHW   65536        // 4*128*128 elements per (b,c)
#define NBC   1024         // B*C = 8*128

// ---------------------------------------------------------------------------
// Stage 1: global max over D,H,W for each (b,c).  One 256-thread block per
// (b,c); each thread strides float4 loads (global_load_b128) over 256 KB.
// ---------------------------------------------------------------------------
__global__ void __launch_bounds__(256) pool_max_kernel(const float* __restrict__ x,
                                                       float* __restrict__ gp) {
    __shared__ float red[256];
    const int bc = blockIdx.x;
    const int t  = threadIdx.x;
    const float4* __restrict__ xv = (const float4*)(x + (size_t)bc * DHW);

    float m = -INFINITY;
    #pragma unroll 4
    for (int j = t; j < DHW / 4; j += 256) {
        float4 v = xv[j];
        m = fmaxf(m, fmaxf(fmaxf(v.x, v.y), fmaxf(v.z, v.w)));
    }
    red[t] = m;
    __syncthreads();
    for (int s = 128; s > 0; s >>= 1) {
        if (t < s) red[t] = fmaxf(red[t], red[t + s]);
        __syncthreads();
    }
    if (t == 0) gp[bc] = red[0];
}

// ---------------------------------------------------------------------------
// Stage 2: tiny MLP gate with V_WMMA_F32_16X16X4_F32.  Single wave32.
//   h = relu(gp @ w1^T + b1) ; g = sigmoid(h @ w2^T + b2)
// Batch rows padded 8 -> 16 (zeros).  A 16x4 f32: lane = M + 16*(K>=2),
// vgpr = K&1.  B 4x16: lane = N + 16*(K>=2), vgpr = K&1.  D 16x16: vgpr r
// holds rows r (lanes 0-15) and r+8 (lanes 16-31).
// ---------------------------------------------------------------------------
__global__ void __launch_bounds__(32) mlp_gate_kernel(const float* __restrict__ gp,
                                                      const float* __restrict__ w1,
                                                      const float* __restrict__ b1,
                                                      const float* __restrict__ w2,
                                                      const float* __restrict__ b2,
                                                      float* __restrict__ g) {
    __shared__ float lgp[16 * CCH];   // padded input activations
    __shared__ float lh[16 * CCH];    // hidden activations
    const int lane  = threadIdx.x;
    const int m     = lane & 15;            // A row / B col / D col for this lane
    const int kh    = (lane >> 4) << 1;     // K-half offset: 0 or 2
    const int half8 = (lane >> 4) << 3;     // D row offset: 0 or 8

    for (int i = lane; i < 16 * CCH; i += 32)
        lgp[i] = (i < 8 * CCH) ? gp[i] : 0.0f;
    __syncthreads();

    // ---- layer 1: h = relu(gp @ w1^T + b1) ----
    for (int nt = 0; nt < 8; ++nt) {
        v8f acc = {};
        const float* __restrict__ arow = lgp + m * CCH;
        const float* __restrict__ wrow = w1 + (size_t)(nt * 16 + m) * CCH;
        #pragma unroll 8
        for (int k0 = 0; k0 < CCH; k0 += 4) {
            v2f a, b;
            a.x = arow[k0 + kh]; a.y = arow[k0 + kh + 1];
            b.x = wrow[k0 + kh]; b.y = wrow[k0 + kh + 1];
            acc = __builtin_amdgcn_wmma_f32_16x16x4_f32(
                false, a, false, b, (short)0, acc, false, false);
        }
        const int col = nt * 16 + m;
        const float bias = b1[col];
        #pragma unroll
        for (int r = 0; r < 8; ++r) {
            float v = acc[r] + bias;
            lh[(r + half8) * CCH + col] = fmaxf(v, 0.0f);
        }
    }
    __syncthreads();

    // ---- layer 2: g = sigmoid(h @ w2^T + b2), keep rows 0..7 ----
    for (int nt = 0; nt < 8; ++nt) {
        v8f acc = {};
        const float* __restrict__ arow = lh + m * CCH;
        const float* __restrict__ wrow = w2 + (size_t)(nt * 16 + m) * CCH;
        #pragma unroll 8
        for (int k0 = 0; k0 < CCH; k0 += 4) {
            v2f a, b;
            a.x = arow[k0 + kh]; a.y = arow[k0 + kh + 1];
            b.x = wrow[k0 + kh]; b.y = wrow[k0 + kh + 1];
            acc = __builtin_amdgcn_wmma_f32_16x16x4_f32(
                false, a, false, b, (short)0, acc, false, false);
        }
        const int col = nt * 16 + m;
        const float bias = b2[col];
        #pragma unroll
        for (int r = 0; r < 8; ++r) {
            const int row = r + half8;
            if (row < 8) {
                float v = acc[r] + bias;
                g[row * CCH + col] = 1.0f / (1.0f + __expf(-v));
            }
        }
    }
}

// ---------------------------------------------------------------------------
// Stage 3: out = x * g[b,c].  Gate index is uniform per block (64 blocks of
// 256 float4 per (b,c), block-aligned) -> scalar load + b128 stream.
// ---------------------------------------------------------------------------
__global__ void __launch_bounds__(256) scale_kernel(const float* __restrict__ x,
                                                    const float* __restrict__ g,
                                                    float* __restrict__ out) {
    const int bc  = blockIdx.x >> 6;        // uniform within block -> SGPR
    const float s = g[bc];
    const size_t i = (size_t)blockIdx.x * 256 + threadIdx.x;
    float4 v = ((const float4*)x)[i];
    float4 r;
    r.x = v.x * s; r.y = v.y * s; r.z = v.z * s; r.w = v.w * s;
    ((float4*)out)[i] = r;
}

// ---------------------------------------------------------------------------
extern "C" void kernel_launch(void* const* d_in, const int* in_sizes, int n_in,
                              void* d_out, int out_size, void* d_ws, size_t ws_size,
                              hipStream_t stream) {
    const float* x  = (const float*)d_in[0];
    const float* w1 = (const float*)d_in[1];
    const float* b1 = (const float*)d_in[2];
    const float* w2 = (const float*)d_in[3];
    const float* b2 = (const float*)d_in[4];
    float* out = (float*)d_out;

    float* gp = (float*)d_ws;       // 1024 floats
    float* g  = gp + NBC;           // 1024 floats

    pool_max_kernel<<<NBC, 256, 0, stream>>>(x, gp);
    mlp_gate_kernel<<<1, 32, 0, stream>>>(gp, w1, b1, w2, b2, g);
    scale_kernel<<<(NBC * DHW) / (4 * 256), 256, 0, stream>>>(x, g, out);
}